// HGTLayer_simplified_16449724744358
// MI455X (gfx1250) — compile-verified
//
#include <hip/hip_runtime.h>

#define LDS_STRIDE 68
#define USE_TDM 1
#define USE_ASYNC_GATHER 1

typedef float v2f __attribute__((ext_vector_type(2)));
typedef float v8f __attribute__((ext_vector_type(8)));
typedef unsigned int v4u __attribute__((ext_vector_type(4)));
typedef unsigned int v8u __attribute__((ext_vector_type(8)));

// ---------------------------------------------------------------------------
// helpers
// ---------------------------------------------------------------------------
__device__ __forceinline__ unsigned lds_off(const void* p) {
  // low 32 bits of a generic shared-memory address are the LDS byte offset
  return (unsigned)(unsigned long long)p;
}
__device__ __forceinline__ void wait_dscnt0() {
  asm volatile("s_wait_dscnt 0x0" ::: "memory");
}
__device__ __forceinline__ void wait_asynccnt0() {
#if USE_ASYNC_GATHER
  asm volatile("s_wait_asynccnt 0x0" ::: "memory");
#endif
}
__device__ __forceinline__ void wait_tensorcnt0() {
#if USE_TDM
  __builtin_amdgcn_s_wait_tensorcnt(0);
#endif
}

__device__ __forceinline__ void atomic_max_float(float* addr, float val) {
  if (val >= 0.0f)
    atomicMax((int*)addr, __float_as_int(val));
  else
    atomicMin((unsigned int*)addr, __float_as_uint(val));
}

// ---------------------------------------------------------------------------
// Weight matrix (64x64 f32, 16KB contiguous) -> LDS with stride-68 rows.
// TDM path: 1-D tile of 4096 dwords, LDS padding 4 dwords every 64 dwords.
// ---------------------------------------------------------------------------
__device__ __forceinline__ void load_weight64(const float* __restrict__ W,
                                              float* Bs, int lane) {
#if USE_TDM
  (void)lane;
  const unsigned long long ga = (unsigned long long)W;
  v4u g0;
  g0[0] = 1u;                                   // count=1, user descriptor
  g0[1] = lds_off(Bs);                          // lds_addr
  g0[2] = (unsigned)ga;                         // global_addr[31:0]
  g0[3] = (2u << 30) | (unsigned)(ga >> 32);    // type=2 | global_addr[56:32]
  v8u g1 = {0u, 0u, 0u, 0u, 0u, 0u, 0u, 0u};
  g1[0] = (2u << 16)        // data_size = 4 bytes
        | (1u << 20)        // pad_enable
        | (5u << 22)        // pad_interval: 64 dwords
        | (3u << 25);       // pad_amount: 4 dwords
  g1[1] = 4096u << 16;      // tensor_dim0[15:0] = 4096
  g1[2] = 1u << 16;         // tensor_dim1[15:0] = 1
  g1[3] = 4096u << 16;      // tile_dim0 = 4096
  g1[4] = 1u;               // tile_dim1 = 1
  g1[5] = 4096u;            // tensor_dim0_stride[31:0]
  g1[6] = 4096u << 16;      // tensor_dim1_stride[15:0]
  asm volatile("tensor_load_to_lds %0, %1" :: "s"(g0), "s"(g1) : "memory");
#else
  #pragma unroll
  for (int i = lane * 4; i < 4096; i += 128) {
    const float4 v = *(const float4*)(W + i);
    *(float4*)(&Bs[(i >> 6) * LDS_STRIDE + (i & 63)]) = v;
  }
#endif
}

// ---------------------------------------------------------------------------
// Per-lane gather of one float4 from a node-feature table into LDS.
// byte_off = id*256 + col*4 (< 2^31). GVS mode: saddr + 32-bit vaddr.
// ---------------------------------------------------------------------------
__device__ __forceinline__ void gather_b128(const float* __restrict__ table,
                                            unsigned byte_off, float* lds_dst) {
#if USE_ASYNC_GATHER
  asm volatile("global_load_async_to_lds_b128 %0, %1, %2"
               :: "v"(lds_off(lds_dst)), "v"(byte_off), "s"(table)
               : "memory");
#else
  *(float4*)lds_dst = *(const float4*)((const char*)table + byte_off);
#endif
}

// ---------------------------------------------------------------------------
// C[16x64] += A[16x64] * B[64x64] via V_WMMA_F32_16X16X4_F32.
// ---------------------------------------------------------------------------
__device__ __forceinline__ void mma_16x64x64(const float* As, const float* Bs,
                                             v8f acc[4], int lane) {
  const int h2 = (lane >> 4) << 1;
  const int ml = lane & 15;
  #pragma unroll
  for (int kk = 0; kk < 16; ++kk) {
    const int k0 = kk * 4 + h2;
    v2f a;
    a.x = As[ml * LDS_STRIDE + k0];
    a.y = As[ml * LDS_STRIDE + k0 + 1];
    #pragma unroll
    for (int nb = 0; nb < 4; ++nb) {
      v2f b;
      b.x = Bs[k0 * LDS_STRIDE + nb * 16 + ml];
      b.y = Bs[(k0 + 1) * LDS_STRIDE + nb * 16 + ml];
      acc[nb] = __builtin_amdgcn_wmma_f32_16x16x4_f32(
          false, a, false, b, (short)0, acc[nb], false, false);
    }
  }
}

// ---------------------------------------------------------------------------
// ctrl layout (ints): [0..7] cnt_type, [8..15] cnt_rel, [16..24] off_type(9),
// [25..33] off_rel(9), [34..41] cur_type, [42..49] cur_rel
// ---------------------------------------------------------------------------

__global__ void hgt_init_kernel(float* __restrict__ tArr, float* __restrict__ mArr,
                                float* __restrict__ den, int* __restrict__ node_ids,
                                int* __restrict__ edge_ids, int* __restrict__ ctrl,
                                int N, int E) {
  const int NT = N * 64;
  const int span = (NT > E + 128) ? NT : (E + 128);
  const int stride = gridDim.x * blockDim.x;
  for (int i = blockIdx.x * blockDim.x + threadIdx.x; i < span; i += stride) {
    if (i < NT) tArr[i] = 0.0f;
    if (i < N) { mArr[i] = __int_as_float(0xff800000); den[i] = 0.0f; }
    if (i < N + 128) node_ids[i] = -1;
    if (i < E + 128) edge_ids[i] = -1;
    if (i < 64) ctrl[i] = 0;
  }
}

__global__ void hgt_count_kernel(const int* __restrict__ ntype,
                                 const int* __restrict__ etype,
                                 int N, int E, int* __restrict__ ctrl) {
  const int span = (N > E) ? N : E;
  const int stride = gridDim.x * blockDim.x;
  for (int i = blockIdx.x * blockDim.x + threadIdx.x; i < span; i += stride) {
    if (i < N) atomicAdd(&ctrl[ntype[i]], 1);
    if (i < E) atomicAdd(&ctrl[8 + etype[i]], 1);
  }
}

__global__ void hgt_scan_kernel(int* __restrict__ ctrl) {
  int o = 0;
  for (int t = 0; t < 8; ++t) { ctrl[16 + t] = o; o += ((ctrl[t] + 15) >> 4) << 4; }
  ctrl[16 + 8] = o;
  o = 0;
  for (int r = 0; r < 8; ++r) { ctrl[25 + r] = o; o += ((ctrl[8 + r] + 15) >> 4) << 4; }
  ctrl[25 + 8] = o;
}

__global__ void hgt_scatter_kernel(const int* __restrict__ ntype,
                                   const int* __restrict__ etype,
                                   int N, int E, int* __restrict__ ctrl,
                                   int* __restrict__ node_ids,
                                   int* __restrict__ edge_ids) {
  const int span = (N > E) ? N : E;
  const int stride = gridDim.x * blockDim.x;
  for (int i = blockIdx.x * blockDim.x + threadIdx.x; i < span; i += stride) {
    if (i < N) {
      const int t = ntype[i];
      const int p = ctrl[16 + t] + atomicAdd(&ctrl[34 + t], 1);
      node_ids[p] = i;
    }
    if (i < E) {
      const int r = etype[i];
      const int p = ctrl[25 + r] + atomicAdd(&ctrl[42 + r], 1);
      edge_ids[p] = i;
    }
  }
}

// ---------------------------------------------------------------------------
// Typed node projection: out[id] = x[id] @ W8[type] (optionally * sigmoid(skip))
// Persistent single-wave blocks over contiguous tile chunks; weight reloaded
// (via TDM) only when the type bin changes.
// ---------------------------------------------------------------------------
__global__ __launch_bounds__(32)
void hgt_proj_kernel(const float* __restrict__ x, const float* __restrict__ W8,
                     const int* __restrict__ node_ids, const int* __restrict__ ctrl,
                     float* __restrict__ out, const float* __restrict__ skip,
                     int do_scale) {
  __shared__ float Bs[64 * LDS_STRIDE];
  __shared__ float As[16 * LDS_STRIDE];
  __shared__ int id_s[16];

  const int lane = threadIdx.x;
  const int* offt = ctrl + 16;
  const int tiles = offt[8] >> 4;
  const int chunk = (tiles + gridDim.x - 1) / gridDim.x;
  int tb = blockIdx.x * chunk;
  int te = tb + chunk; if (te > tiles) te = tiles;
  if (tb >= te) return;

  const int h = lane >> 4, ml = lane & 15;
  int curT = -1;
  for (int tile = tb; tile < te; ++tile) {
    const int base = tile << 4;
    int t = 0;
    #pragma unroll
    for (int i = 1; i < 8; ++i) t += (base >= offt[i]) ? 1 : 0;

    wait_dscnt0();  // previous tile's LDS reads done before As/Bs are rewritten
    if (t != curT) { load_weight64(W8 + (size_t)t * 4096, Bs, lane); curT = t; }
    if (lane < 16) id_s[lane] = node_ids[base + lane];
    __syncthreads();

    #pragma unroll
    for (int j = lane; j < 256; j += 32) {
      const int row = j >> 4, c4 = (j & 15) << 2;
      int id = id_s[row]; if (id < 0) id = 0;  // sentinel row: harmless garbage
      gather_b128(x, (unsigned)(id * 256 + c4 * 4), &As[row * LDS_STRIDE + c4]);
    }
    wait_asynccnt0();
    wait_tensorcnt0();
    __syncthreads();

    const v8f zero = {0.f, 0.f, 0.f, 0.f, 0.f, 0.f, 0.f, 0.f};
    v8f acc[4] = {zero, zero, zero, zero};
    mma_16x64x64(As, Bs, acc, lane);

    float alpha = 1.0f;
    if (do_scale) alpha = 1.0f / (1.0f + __expf(-skip[t]));

    #pragma unroll
    for (int nb = 0; nb < 4; ++nb) {
      #pragma unroll
      for (int i = 0; i < 8; ++i) {
        const int row = i + 8 * h;
        const int id = id_s[row];
        if (id >= 0) out[(size_t)id * 64 + nb * 16 + ml] = acc[nb][i] * alpha;
      }
    }
  }
}

// ---------------------------------------------------------------------------
// Pass A: k_rel = k_n[src] @ rel_att[r]; attn = <k_rel, q_n[dst]> * pri / 8
// plus atomic segment-max into mArr[dst].
// ---------------------------------------------------------------------------
__global__ __launch_bounds__(32)
void hgt_edge_attn_kernel(const float* __restrict__ kn, const float* __restrict__ qn,
                          const float* __restrict__ rel_att,
                          const float* __restrict__ rel_pri,
                          const int* __restrict__ src, const int* __restrict__ dst,
                          const int* __restrict__ edge_ids, const int* __restrict__ ctrl,
                          float* __restrict__ attn, float* __restrict__ mArr) {
  __shared__ float Bs[64 * LDS_STRIDE];
  __shared__ float As[16 * LDS_STRIDE];
  __shared__ float Qs[16 * LDS_STRIDE];
  __shared__ float Ks[16 * LDS_STRIDE];
  __shared__ float part[32];
  __shared__ int eid_s[16], src_s[16], dst_s[16];

  const int lane = threadIdx.x;
  const int* offr = ctrl + 25;
  const int tiles = offr[8] >> 4;
  const int chunk = (tiles + gridDim.x - 1) / gridDim.x;
  int tb = blockIdx.x * chunk;
  int te = tb + chunk; if (te > tiles) te = tiles;
  if (tb >= te) return;

  const int h = lane >> 4, ml = lane & 15;
  int curR = -1;
  for (int tile = tb; tile < te; ++tile) {
    const int base = tile << 4;
    int r = 0;
    #pragma unroll
    for (int i = 1; i < 8; ++i) r += (base >= offr[i]) ? 1 : 0;

    wait_dscnt0();
    if (r != curR) { load_weight64(rel_att + (size_t)r * 4096, Bs, lane); curR = r; }
    if (lane < 16) {
      const int e = edge_ids[base + lane];
      eid_s[lane] = e;
      src_s[lane] = (e >= 0) ? src[e] : 0;
      dst_s[lane] = (e >= 0) ? dst[e] : 0;
    }
    __syncthreads();

    #pragma unroll
    for (int j = lane; j < 256; j += 32) {
      const int row = j >> 4, c4 = (j & 15) << 2;
      const unsigned co = (unsigned)(c4 * 4);
      gather_b128(kn, (unsigned)(src_s[row] * 256) + co, &As[row * LDS_STRIDE + c4]);
      gather_b128(qn, (unsigned)(dst_s[row] * 256) + co, &Qs[row * LDS_STRIDE + c4]);
    }
    wait_asynccnt0();
    wait_tensorcnt0();
    __syncthreads();

    const v8f zero = {0.f, 0.f, 0.f, 0.f, 0.f, 0.f, 0.f, 0.f};
    v8f acc[4] = {zero, zero, zero, zero};
    mma_16x64x64(As, Bs, acc, lane);

    #pragma unroll
    for (int nb = 0; nb < 4; ++nb) {
      #pragma unroll
      for (int i = 0; i < 8; ++i)
        Ks[(i + 8 * h) * LDS_STRIDE + nb * 16 + ml] = acc[nb][i];
    }
    __syncthreads();

    float p = 0.f;
    #pragma unroll
    for (int c = 0; c < 32; ++c) {
      const int col = 32 * h + c;
      p += Ks[ml * LDS_STRIDE + col] * Qs[ml * LDS_STRIDE + col];
    }
    part[lane] = p;
    __syncthreads();

    if (lane < 16) {
      const int e = eid_s[lane];
      if (e >= 0) {
        const float av = (part[lane] + part[lane + 16]) * rel_pri[r] * 0.125f;
        attn[e] = av;
        atomic_max_float(&mArr[dst_s[lane]], av);
      }
    }
  }
}

// ---------------------------------------------------------------------------
// Pass B: ex = exp(attn - m[dst]); den[dst] += ex; attn <- ex
// ---------------------------------------------------------------------------
__global__ void hgt_softmax_kernel(const int* __restrict__ dst,
                                   float* __restrict__ attn,
                                   const float* __restrict__ mArr,
                                   float* __restrict__ den, int E) {
  const int e = blockIdx.x * blockDim.x + threadIdx.x;
  if (e >= E) return;
  const int d = dst[e];
  const float ex = __expf(attn[e] - mArr[d]);
  attn[e] = ex;
  atomicAdd(&den[d], ex);
}

// ---------------------------------------------------------------------------
// Pass C: v_rel = v_n[src] @ rel_msg[r]; t[dst] += (ex/den[dst]) * v_rel
// ---------------------------------------------------------------------------
__global__ __launch_bounds__(32)
void hgt_edge_aggr_kernel(const float* __restrict__ vn,
                          const float* __restrict__ rel_msg,
                          const int* __restrict__ src, const int* __restrict__ dst,
                          const int* __restrict__ edge_ids, const int* __restrict__ ctrl,
                          const float* __restrict__ attn_ex,
                          const float* __restrict__ den,
                          float* __restrict__ tArr) {
  __shared__ float Bs[64 * LDS_STRIDE];
  __shared__ float As[16 * LDS_STRIDE];
  __shared__ float w_s[16];
  __shared__ int eid_s[16], src_s[16], dst_s[16];

  const int lane = threadIdx.x;
  const int* offr = ctrl + 25;
  const int tiles = offr[8] >> 4;
  const int chunk = (tiles + gridDim.x - 1) / gridDim.x;
  int tb = blockIdx.x * chunk;
  int te = tb + chunk; if (te > tiles) te = tiles;
  if (tb >= te) return;

  const int h = lane >> 4, ml = lane & 15;
  int curR = -1;
  for (int tile = tb; tile < te; ++tile) {
    const int base = tile << 4;
    int r = 0;
    #pragma unroll
    for (int i = 1; i < 8; ++i) r += (base >= offr[i]) ? 1 : 0;

    wait_dscnt0();
    if (r != curR) { load_weight64(rel_msg + (size_t)r * 4096, Bs, lane); curR = r; }
    if (lane < 16) {
      const int e = edge_ids[base + lane];
      eid_s[lane] = e;
      src_s[lane] = (e >= 0) ? src[e] : 0;
      const int d = (e >= 0) ? dst[e] : 0;
      dst_s[lane] = d;
      w_s[lane] = (e >= 0) ? attn_ex[e] / den[d] : 0.0f;
    }
    __syncthreads();

    #pragma unroll
    for (int j = lane; j < 256; j += 32) {
      const int row = j >> 4, c4 = (j & 15) << 2;
      gather_b128(vn, (unsigned)(src_s[row] * 256 + c4 * 4),
                  &As[row * LDS_STRIDE + c4]);
    }
    wait_asynccnt0();
    wait_tensorcnt0();
    __syncthreads();

    const v8f zero = {0.f, 0.f, 0.f, 0.f, 0.f, 0.f, 0.f, 0.f};
    v8f acc[4] = {zero, zero, zero, zero};
    mma_16x64x64(As, Bs, acc, lane);

    #pragma unroll
    for (int nb = 0; nb < 4; ++nb) {
      #pragma unroll
      for (int i = 0; i < 8; ++i) {
        const int row = i + 8 * h;
        if (eid_s[row] >= 0)
          atomicAdd(&tArr[(size_t)dst_s[row] * 64 + nb * 16 + ml],
                    acc[nb][i] * w_s[row]);
      }
    }
  }
}

// ---------------------------------------------------------------------------
extern "C" void kernel_launch(void* const* d_in, const int* in_sizes, int n_in,
                              void* d_out, int out_size, void* d_ws, size_t ws_size,
                              hipStream_t stream) {
  (void)n_in; (void)out_size; (void)ws_size;
  const float* h        = (const float*)d_in[0];
  const float* k_w      = (const float*)d_in[1];
  const float* q_w      = (const float*)d_in[2];
  const float* v_w      = (const float*)d_in[3];
  const float* a_w      = (const float*)d_in[4];
  const float* rel_att  = (const float*)d_in[5];
  const float* rel_msg  = (const float*)d_in[6];
  const float* rel_pri  = (const float*)d_in[7];
  const float* skip     = (const float*)d_in[8];
  const int*   src      = (const int*)d_in[9];
  const int*   dst      = (const int*)d_in[10];
  const int*   etype    = (const int*)d_in[11];
  const int*   ntype    = (const int*)d_in[12];
  const int N = in_sizes[12];
  const int E = in_sizes[9];

  char* ws = (char*)d_ws;
  size_t cur = 0;
  auto take = [&](size_t bytes) -> void* {
    cur = (cur + 255) & ~(size_t)255;
    void* p = ws + cur;
    cur += bytes;
    return p;
  };
  float* kn       = (float*)take((size_t)N * 64 * 4);
  float* qn       = (float*)take((size_t)N * 64 * 4);
  float* vn       = (float*)take((size_t)N * 64 * 4);
  float* tArr     = (float*)take((size_t)N * 64 * 4);
  float* mArr     = (float*)take((size_t)N * 4);
  float* den      = (float*)take((size_t)N * 4);
  float* attn     = (float*)take((size_t)E * 4);
  int*   node_ids = (int*)take((size_t)(N + 128) * 4);
  int*   edge_ids = (int*)take((size_t)(E + 128) * 4);
  int*   ctrl     = (int*)take(64 * 4);

  hgt_init_kernel<<<4096, 256, 0, stream>>>(tArr, mArr, den, node_ids, edge_ids,
                                            ctrl, N, E);
  hgt_count_kernel<<<2048, 256, 0, stream>>>(ntype, etype, N, E, ctrl);
  hgt_scan_kernel<<<1, 1, 0, stream>>>(ctrl);
  hgt_scatter_kernel<<<2048, 256, 0, stream>>>(ntype, etype, N, E, ctrl,
                                               node_ids, edge_ids);

  hgt_proj_kernel<<<1024, 32, 0, stream>>>(h, k_w, node_ids, ctrl, kn, nullptr, 0);
  hgt_proj_kernel<<<1024, 32, 0, stream>>>(h, q_w, node_ids, ctrl, qn, nullptr, 0);
  hgt_proj_kernel<<<1024, 32, 0, stream>>>(h, v_w, node_ids, ctrl, vn, nullptr, 0);

  hgt_edge_attn_kernel<<<4096, 32, 0, stream>>>(kn, qn, rel_att, rel_pri,
                                                src, dst, edge_ids, ctrl,
                                                attn, mArr);
  hgt_softmax_kernel<<<(E + 255) / 256, 256, 0, stream>>>(dst, attn, mArr, den, E);
  hgt_edge_aggr_kernel<<<4096, 32, 0, stream>>>(vn, rel_msg, src, dst,
                                                edge_ids, ctrl, attn, den, tArr);

  hgt_proj_kernel<<<1024, 32, 0, stream>>>(tArr, a_w, node_ids, ctrl,
                                           (float*)d_out, skip, 1);
}